// DiffVolume_85048942395515
// MI455X (gfx1250) — compile-verified
//
#include <hip/hip_runtime.h>

// Problem dims (fixed by the reference's setup_inputs)
#define BB 2
#define CC 32
#define HH 96
#define WW 320
#define DD 48

typedef __attribute__((ext_vector_type(4))) float v4f;
typedef __attribute__((ext_vector_type(4))) int   v4i;

// gfx1250 async global->LDS path (ASYNCcnt), with portable fallback.
// Guards also keep the host compilation pass happy (builtins absent on x86).
#if defined(__gfx1250__) &&                                                   \
    __has_builtin(__builtin_amdgcn_global_load_async_to_lds_b128) &&          \
    __has_builtin(__builtin_amdgcn_s_wait_asynccnt)
#define USE_ASYNC_LDS 1
#else
#define USE_ASYNC_LDS 0
#endif

__global__ __launch_bounds__(160) void
diff_volume_kernel(const float* __restrict__ left,
                   const float* __restrict__ right,
                   float* __restrict__ out)
{
    // One block per (b*c, h) row. 2*320 floats = 2.5 KB LDS.
    __shared__ alignas(16) float l_lds[WW];
    __shared__ alignas(16) float r_lds[WW];

    const int tid = threadIdx.x;      // 0..159 (5 wave32s)
    const int row = blockIdx.x;       // 0..B*C*H-1
    const int h   = row % HH;
    const int bc  = row / HH;

    const float* __restrict__ lrow = left  + (size_t)(bc * HH + h) * WW;
    const float* __restrict__ rrow = right + (size_t)(bc * HH + h) * WW;

    // ---- Stage both rows into LDS: threads 0..79 -> left, 80..159 -> right.
    {
        const int  isR = (tid >= 80);
        const int  q   = isR ? (tid - 80) : tid;          // float4 chunk 0..79
        const float* gsrc = (isR ? rrow : lrow) + 4 * q;
        float*       ldst = (isR ? r_lds : l_lds) + 4 * q;
#if USE_ASYNC_LDS
        __builtin_amdgcn_global_load_async_to_lds_b128(
            (v4i*)gsrc, (v4i*)ldst, /*offset=*/0, /*cpol=*/0);
        __builtin_amdgcn_s_wait_asynccnt(0);
#else
        *(v4f*)ldst = *(const v4f*)gsrc;
#endif
    }
    __syncthreads();

    // ---- Compute: each thread owns one 16B output chunk (w0..w0+3),
    //      iterating disparities d = d0, d0+2, ... (24 iterations).
    const int q  = (tid < 80) ? tid : (tid - 80);
    const int d0 = (tid < 80) ? 0 : 1;
    const int w0 = 4 * q;

    const v4f lv = *(const v4f*)(l_lds + w0);   // left values: loop-invariant

    // out[((bc*D + d)*H + h)*W + w]
    float* __restrict__ obase = out + ((size_t)bc * DD * HH + h) * WW + w0;

#pragma unroll 4
    for (int i = 0; i < 24; ++i) {
        const int d = d0 + 2 * i;
        v4f o;
#pragma unroll
        for (int j = 0; j < 4; ++j) {
            const int w   = w0 + j;
            const int src = w - d;
            const float rv = r_lds[src < 0 ? 0 : src];   // clamped LDS gather
            o[j] = (w >= d) ? (lv[j] - rv) : 0.0f;
        }
        // Output is never re-read: non-temporal b128 store, coalesced per wave.
        __builtin_nontemporal_store(o, (v4f*)(obase + (size_t)d * HH * WW));
    }
}

extern "C" void kernel_launch(void* const* d_in, const int* in_sizes, int n_in,
                              void* d_out, int out_size, void* d_ws, size_t ws_size,
                              hipStream_t stream)
{
    (void)in_sizes; (void)n_in; (void)out_size; (void)d_ws; (void)ws_size;
    const float* left  = (const float*)d_in[0];
    const float* right = (const float*)d_in[1];
    float*       out   = (float*)d_out;

    const int nblocks = BB * CC * HH;   // 6144 row-blocks
    diff_volume_kernel<<<nblocks, 160, 0, stream>>>(left, right, out);
}